// RoIPool_56040733278796
// MI455X (gfx1250) — compile-verified
//
#include <hip/hip_runtime.h>
#include <stdint.h>

// RoI max-pool for MI455X (gfx1250), TDM-pipelined version.
//
// Roofline: output 6.4 MB write; features 10.2 MB are L2-resident (192 MB L2),
// so ideal time ~ (10 MB first-touch read + 6.4 MB write)/23.3 TB/s ~ 0.7 us.
// Pure data movement -> no WMMA (max-plus semiring, not matmul). The CDNA5
// feature that fits is the Tensor Data Mover: each (roi, channel) needs a 2D
// sub-tile (roi_h x roi_w, stride W) of a row-major tensor DMA'd into LDS.
//
// Pipeline: one wave32 owns one roi and CPW=8 consecutive channels with two
// LDS tile buffers. Iteration k issues the DMA for channel k+1 into the
// alternate buffer, then s_wait_tensorcnt 1 (TDM ops from one wave complete
// in order => channel k's tile is resident), computes the 7x7 bin maxes from
// LDS and stores 49 contiguous floats. DMA latency is overlapped with compute
// inside each wave, on top of 4096-wave occupancy.

#define POOLED_HW   7
#define SPATIAL_SC  0.0625f
#define FEAT_C      256
#define FEAT_H      50
#define FEAT_W      50
#define NROI        128
#define WAVES_PB    2
#define CPW         8                    // channels per wave (pipelined DMAs)
#define CGRPS       (FEAT_C / CPW)       // 32 channel-groups per roi
#define TILE_F      2604                 // 51*51 worst-case tile, padded to x4

typedef __attribute__((ext_vector_type(4))) unsigned int v4u;
typedef __attribute__((ext_vector_type(8))) int          v8i;
typedef __attribute__((ext_vector_type(4))) int          v4i;

#if __has_builtin(__builtin_amdgcn_tensor_load_to_lds)
#define HAVE_TDM 1
#endif

#ifdef HAVE_TDM
// Issue one 2D-tile TDM load: tile_h x tile_w fp32, row stride FEAT_W,
// OOB (>= tdim0/tdim1 from the tile origin) reads return 0.
// Descriptor bit layout per CDNA5 ISA 8.3/8.4.
__device__ __forceinline__ void tdm_issue_2d(uint32_t ldsoff, uint64_t gaddr,
                                             uint32_t tdim0, uint32_t tdim1,
                                             uint32_t tile_w, uint32_t tile_h) {
    v4u g0;
    g0[0] = 1u;                                       // count=1 (valid), user mode
    g0[1] = ldsoff;                                   // lds_addr
    g0[2] = (uint32_t)gaddr;                          // global_addr[31:0]
    g0[3] = (uint32_t)(gaddr >> 32) | (2u << 30);     // global_addr[56:32], type=2
    v8i g1;
    g1[0] = (int)(2u << 16);                          // wg_mask=0, data_size=4B
    g1[1] = (int)((tdim0 & 0xFFFFu) << 16);           // barrier_addr=0 | tensor_dim0 lo
    g1[2] = (int)((tdim0 >> 16) | ((tdim1 & 0xFFFFu) << 16)); // dim0 hi | dim1 lo
    g1[3] = (int)((tdim1 >> 16) | (tile_w << 16));    // dim1 hi | tile_dim0
    g1[4] = (int)tile_h;                              // tile_dim1, tile_dim2=0
    g1[5] = FEAT_W;                                   // tensor_dim0_stride[31:0]
    g1[6] = 0;                                        // stride0 hi | stride1 lo
    g1[7] = 0;                                        // stride1 hi
    v4i gz = {0, 0, 0, 0};                            // groups 2/3 unused (2D tile)
#if defined(__clang_major__) && (__clang_major__ >= 23)
    v8i gz8 = {0, 0, 0, 0, 0, 0, 0, 0};
    __builtin_amdgcn_tensor_load_to_lds(g0, g1, gz, gz, gz8, 0);
#else
    __builtin_amdgcn_tensor_load_to_lds(g0, g1, gz, gz, 0);
#endif
}
#endif

// 7x7 bin maxes for one channel tile resident in LDS; lanes split the 49 bins.
__device__ __forceinline__ void pool_bins(const float* __restrict__ t,
                                          int lane, int x1, int y1,
                                          int roi_w, float bin_w, float bin_h,
                                          float* __restrict__ op) {
    for (int bin = lane; bin < POOLED_HW * POOLED_HW; bin += 32) {
        const int ph = bin / POOLED_HW;
        const int pw = bin - ph * POOLED_HW;
        int hs = min((int)floorf((float)ph       * bin_h) + y1, FEAT_H);
        int he = min((int)ceilf ((float)(ph + 1) * bin_h) + y1, FEAT_H);
        int ws = min((int)floorf((float)pw       * bin_w) + x1, FEAT_W);
        int we = min((int)ceilf ((float)(pw + 1) * bin_w) + x1, FEAT_W);
        hs = max(hs, 0); ws = max(ws, 0);
        float m = 0.0f;                               // empty bin -> 0
        if (he > hs && we > ws) {
            m = -__builtin_inff();
            for (int h = hs; h < he; ++h) {
                const float* row = t + (h - y1) * roi_w;
                for (int w = ws; w < we; ++w)
                    m = fmaxf(m, row[w - x1]);
            }
        }
        op[bin] = m;
    }
}

__global__ __launch_bounds__(WAVES_PB * 32)
void RoIPool_56040733278796_kernel(const float* __restrict__ feat,
                                   const float* __restrict__ rois,
                                   float* __restrict__ out) {
    __shared__ float tile[WAVES_PB][2][TILE_F];       // double buffer per wave

    const int lane  = threadIdx.x & 31;
    const int wv    = threadIdx.x >> 5;
    const int wid   = blockIdx.x * WAVES_PB + wv;     // wave = (roi, channel-group)
    const int r     = wid >> 5;                       // / CGRPS (32, power of 2)
    const int cbase = (wid & (CGRPS - 1)) * CPW;

    // ---- RoI parameters (identical across the wave) ----
    const float* rp = rois + (size_t)r * 5;
    int b  = (int)rp[0];
    int x1 = (int)floorf(rp[1] * SPATIAL_SC + 0.5f);
    int y1 = (int)floorf(rp[2] * SPATIAL_SC + 0.5f);
    int x2 = (int)floorf(rp[3] * SPATIAL_SC + 0.5f);
    int y2 = (int)floorf(rp[4] * SPATIAL_SC + 0.5f);
    int roi_w = max(x2 - x1 + 1, 1);                  // <= 51
    int roi_h = max(y2 - y1 + 1, 1);                  // <= 51
    // Force wave-uniform so the TDM descriptor lives in SGPRs.
    b     = __builtin_amdgcn_readfirstlane(b);
    x1    = __builtin_amdgcn_readfirstlane(x1);
    y1    = __builtin_amdgcn_readfirstlane(y1);
    roi_w = __builtin_amdgcn_readfirstlane(roi_w);
    roi_h = __builtin_amdgcn_readfirstlane(roi_h);

    const float bin_w = (float)roi_w * (1.0f / 7.0f);
    const float bin_h = (float)roi_h * (1.0f / 7.0f);
    const uint32_t tdim0 = (uint32_t)max(FEAT_W - x1, 0);  // OOB cols -> 0 (unused)
    const uint32_t tdim1 = (uint32_t)max(FEAT_H - y1, 0);  // OOB rows -> 0 (unused)

    // Channel-k tile origin: feat[b][cbase+k][y1][x1]; +H*W floats per channel.
    const uint64_t gbase = (uint64_t)(uintptr_t)
        (feat + (((size_t)b * FEAT_C + cbase) * FEAT_H + (size_t)y1) * FEAT_W + (size_t)x1);
    const uint64_t gstep = (uint64_t)FEAT_H * FEAT_W * sizeof(float);
    const uint32_t lds0  = (uint32_t)(uintptr_t)&tile[wv][0][0];  // WG-relative offset
    const uint32_t lds1  = (uint32_t)(uintptr_t)&tile[wv][1][0];
    float* op = out + (((size_t)r * FEAT_C + cbase) * (POOLED_HW * POOLED_HW));

#ifdef HAVE_TDM
    // ---- Software-pipelined TDM double buffer ----
    tdm_issue_2d(lds0, gbase, tdim0, tdim1, (uint32_t)roi_w, (uint32_t)roi_h);
    for (int k = 0; k < CPW; ++k) {
        if (k + 1 < CPW) {
            // Prior LDS reads of the buffer we are about to overwrite must
            // retire first: TDM writes are unordered with this wave's DS ops.
            asm volatile("s_wait_dscnt 0x0" ::: "memory");
            tdm_issue_2d((k & 1) ? lds0 : lds1, gbase + (uint64_t)(k + 1) * gstep,
                         tdim0, tdim1, (uint32_t)roi_w, (uint32_t)roi_h);
#if __has_builtin(__builtin_amdgcn_s_wait_tensorcnt)
            __builtin_amdgcn_s_wait_tensorcnt(1);     // in-order => tile k resident
#else
            asm volatile("s_wait_tensorcnt 0x1" ::: "memory");
#endif
        } else {
#if __has_builtin(__builtin_amdgcn_s_wait_tensorcnt)
            __builtin_amdgcn_s_wait_tensorcnt(0);
#else
            asm volatile("s_wait_tensorcnt 0x0" ::: "memory");
#endif
        }
        asm volatile("" ::: "memory");                // no LDS reads hoisted above wait
        pool_bins(&tile[wv][k & 1][0], lane, x1, y1, roi_w, bin_w, bin_h,
                  op + (size_t)k * (POOLED_HW * POOLED_HW));
    }
#else
    // ---- Fallback: cooperative lane loads, one channel at a time ----
    for (int k = 0; k < CPW; ++k) {
        const float* src = (const float*)(uintptr_t)(gbase + (uint64_t)k * gstep);
        const int n = roi_w * roi_h;
        for (int i = lane; i < n; i += 32) {
            const int ty = i / roi_w, tx = i - ty * roi_w;
            float v = 0.0f;
            if ((uint32_t)tx < tdim0 && (uint32_t)ty < tdim1)
                v = src[(size_t)ty * FEAT_W + tx];
            tile[wv][0][i] = v;
        }
        __builtin_amdgcn_wave_barrier();
        pool_bins(&tile[wv][0][0], lane, x1, y1, roi_w, bin_w, bin_h,
                  op + (size_t)k * (POOLED_HW * POOLED_HW));
        __builtin_amdgcn_wave_barrier();
    }
#endif
}

extern "C" void kernel_launch(void* const* d_in, const int* in_sizes, int n_in,
                              void* d_out, int out_size, void* d_ws, size_t ws_size,
                              hipStream_t stream) {
    (void)in_sizes; (void)n_in; (void)out_size; (void)d_ws; (void)ws_size;
    const float* feat = (const float*)d_in[0];
    const float* rois = (const float*)d_in[1];
    float* out = (float*)d_out;
    const int total_waves = NROI * CGRPS;             // 4096 waves, 8 DMAs each
    const int blocks = total_waves / WAVES_PB;        // 2048 blocks of 2 waves
    RoIPool_56040733278796_kernel<<<blocks, WAVES_PB * 32, 0, stream>>>(feat, rois, out);
}